// OpticalConv2dNew_63668595196003
// MI455X (gfx1250) — compile-verified
//
#include <hip/hip_runtime.h>

// ---------------------------------------------------------------------------
// OpticalConv2d == circular 3x3 conv of the channel-summed image with
// kernel[:,0], done as a per-batch GEMM  Out[256x784] = W[256x9] x S[9x784]
// on v_wmma_f32_16x16x4_f32 (K padded 9 -> 12 = 3 x K=4 WMMA steps).
//
// Memory-bound: read x once (205 MB) + write out once (205 MB) @ 23.3 TB/s
// => ~18 us floor. Strategy: single pass over x, xs[b] staged in LDS (3.1 KB),
// weights in VGPRs per wave, NT hints on both streaming tensors, two output
// tiles in flight per wave to cover WMMA->WMMA RAW hazard bubbles.
// ---------------------------------------------------------------------------

typedef __attribute__((ext_vector_type(2))) float v2f;
typedef __attribute__((ext_vector_type(8))) float v8f;

#define NSIDE  28
#define NPIX   784     // 28*28
#define CIN    256
#define COUT   256
#define NBATCH 256
#define KTAPS  9       // 3x3 taps, K padded to 12

// Build the B-matrix fragment (3 x K=4 steps) for one 16-pixel tile.
// Lane layout for v_wmma_f32_16x16x4_f32 B: col N = lane%16;
// VGPR0 = K(2*hi), VGPR1 = K(2*hi+1) with hi = lane>>4.
__device__ __forceinline__ void make_b(const float* __restrict__ xs,
                                       int t, int m, int hi, v2f bv[3]) {
  const int p = t * 16 + m;
  const int i = p / NSIDE;
  const int j = p - i * NSIDE;
  #pragma unroll
  for (int s = 0; s < 3; ++s) {
    float vals[2];
    #pragma unroll
    for (int q = 0; q < 2; ++q) {
      const int k  = 4 * s + 2 * hi + q;
      const int kc = (k < KTAPS) ? k : 0;         // selects, no divergence
      const int u  = kc / 3;
      const int v  = kc - 3 * u;
      int ii = i + 2 - u; if (ii >= NSIDE) ii -= NSIDE;   // circular shift
      int jj = j + 2 - v; if (jj >= NSIDE) jj -= NSIDE;
      vals[q] = xs[ii * NSIDE + jj] * ((k < KTAPS) ? 1.0f : 0.0f);
    }
    bv[s].x = vals[0];
    bv[s].y = vals[1];
  }
}

__launch_bounds__(512, 1)
__global__ void optical_conv_wmma(const float* __restrict__ x,
                                  const float* __restrict__ kern,
                                  float* __restrict__ out) {
  __shared__ float xs[NPIX];           // channel-summed 28x28 image, 3.1 KB

  const int b   = blockIdx.x;
  const int tid = threadIdx.x;

  // ---------- Phase A: xs[p] = sum_c x[b,c,p] (coalesced, NT loads) ---------
  const float* xb = x + (size_t)b * CIN * NPIX;
  for (int p = tid; p < NPIX; p += 512) {
    float acc = 0.0f;
    #pragma unroll 8
    for (int c = 0; c < CIN; ++c)
      acc += __builtin_nontemporal_load(&xb[c * NPIX + p]);
    xs[p] = acc;
  }
  __syncthreads();

  // ---------- Phase B: WMMA tiles -------------------------------------------
  // wave w (of 16) owns couts [16w, 16w+16); loops 49 pixel tiles of 16.
  const int wave = tid >> 5;           // 0..15
  const int lane = tid & 31;
  const int m    = lane & 15;          // A-row (cout) / B-col (pixel) in tile
  const int hi   = lane >> 4;          // K-pair select
  const int coutBase = wave * 16;

  // A: W[m,k] = kern[(coutBase+m)*CIN*9 + k] (ci = 0 slice), k in [0,9), pad 0.
  v2f a[3];
  const int arow = (coutBase + m) * (CIN * KTAPS);
  #pragma unroll
  for (int s = 0; s < 3; ++s) {
    const int k0  = 4 * s + 2 * hi;
    const int k1  = k0 + 1;
    const int k0c = (k0 < KTAPS) ? k0 : 0;
    const int k1c = (k1 < KTAPS) ? k1 : 0;
    a[s].x = kern[arow + k0c] * ((k0 < KTAPS) ? 1.0f : 0.0f);
    a[s].y = kern[arow + k1c] * ((k1 < KTAPS) ? 1.0f : 0.0f);
  }

  float* ob = out + ((size_t)b * COUT + coutBase) * NPIX;

  // Two tiles in flight: independent accumulator chains hide WMMA RAW hazards.
  for (int t = 0; t < 48; t += 2) {
    v2f bv0[3], bv1[3];
    make_b(xs, t,     m, hi, bv0);
    make_b(xs, t + 1, m, hi, bv1);

    v8f acc0 = {}, acc1 = {};
    #pragma unroll
    for (int s = 0; s < 3; ++s) {
      acc0 = __builtin_amdgcn_wmma_f32_16x16x4_f32(
          false, a[s], false, bv0[s], (short)0, acc0, false, false);
      acc1 = __builtin_amdgcn_wmma_f32_16x16x4_f32(
          false, a[s], false, bv1[s], (short)0, acc1, false, false);
    }

    // D layout: VGPR r, lane -> cout = coutBase + r + 8*hi, pixel = t*16 + m
    #pragma unroll
    for (int r = 0; r < 8; ++r) {
      const size_t row = (size_t)(r + 8 * hi) * NPIX;
      __builtin_nontemporal_store(acc0[r], &ob[row + (t    ) * 16 + m]);
      __builtin_nontemporal_store(acc1[r], &ob[row + (t + 1) * 16 + m]);
    }
  }

  // Tail tile t = 48
  {
    const int t = 48;
    v2f bv[3];
    make_b(xs, t, m, hi, bv);
    v8f acc = {};
    #pragma unroll
    for (int s = 0; s < 3; ++s) {
      acc = __builtin_amdgcn_wmma_f32_16x16x4_f32(
          false, a[s], false, bv[s], (short)0, acc, false, false);
    }
    #pragma unroll
    for (int r = 0; r < 8; ++r) {
      __builtin_nontemporal_store(
          acc[r], &ob[(size_t)(r + 8 * hi) * NPIX + t * 16 + m]);
    }
  }
}

extern "C" void kernel_launch(void* const* d_in, const int* in_sizes, int n_in,
                              void* d_out, int out_size, void* d_ws, size_t ws_size,
                              hipStream_t stream) {
  const float* x    = (const float*)d_in[0];   // [256,256,28,28] f32
  const float* kern = (const float*)d_in[1];   // [256,256,3,3]  f32 (only ci=0 used)
  float* out        = (float*)d_out;           // [256,256,28,28] f32

  (void)in_sizes; (void)n_in; (void)out_size; (void)d_ws; (void)ws_size;

  optical_conv_wmma<<<NBATCH, 512, 0, stream>>>(x, kern, out);
}